// ChartParser_44092134261529
// MI455X (gfx1250) — compile-verified
//
#include <hip/hip_runtime.h>

#define L_SEQ 384
#define HD 1024
#define HALF 512
#define NLAB 128
#define NSPANS ((L_SEQ * (L_SEQ + 1)) / 2) /* 73920 */

#define PK 128   /* K panel size */
#define PKP 132  /* padded row stride (floats): 528 B = 16B-aligned, banks spread */
#define WPB 4    /* waves per block in scores kernel */
#define NBLK (NSPANS / (16 * WPB)) /* 1155, exact */

typedef __attribute__((ext_vector_type(2))) float v2f;
typedef __attribute__((ext_vector_type(8))) float v8f;

// ---------------------------------------------------------------------------
// Kernel 1: per-span (length,left) tables in triangle layout (rows by length),
// identical enumeration to the reference's _span_indices().
// ---------------------------------------------------------------------------
__global__ void spans_init_kernel(int* __restrict__ spanL, int* __restrict__ spanLeft) {
  int l = blockIdx.x * blockDim.x + threadIdx.x + 1; // 1..L
  if (l > L_SEQ) return;
  int off = (l - 1) * L_SEQ - ((l - 1) * (l - 2)) / 2;
  int N = L_SEQ - l + 1;
  for (int i = 0; i < N; ++i) {
    spanL[off + i] = l;
    spanLeft[off + i] = i;
  }
}

// ---------------------------------------------------------------------------
// Kernel 2: row projections PF[i][j] = sum_k Hf[i][k]*W1[k][j],
//           PB[i][j] = sum_k Hb[i][k]*W1[512+k][j].  (386x512x1024) x2, tiny
// (0.8 GFLOP) thanks to linearity of feats@W1 in the hidden rows.
// ---------------------------------------------------------------------------
__global__ void proj_kernel(const float* __restrict__ hiddens, const float* __restrict__ W1,
                            float* __restrict__ PF, float* __restrict__ PB) {
  int idx = blockIdx.x * blockDim.x + threadIdx.x; // 2*(L+2)*1024 total
  int j = idx & (HD - 1);
  int t = idx >> 10;
  int row = t >> 1;
  int half = t & 1;
  if (row >= L_SEQ + 2) return;
  const float* h = hiddens + (size_t)row * HD + half * HALF;
  const float* w = W1 + (size_t)half * HALF * HD + j;
  float acc = 0.f;
#pragma unroll 8
  for (int k = 0; k < HALF; ++k) acc = fmaf(h[k], w[(size_t)k * HD], acc);
  float* dst = half ? PB : PF;
  dst[(size_t)row * HD + j] = acc;
}

// ---------------------------------------------------------------------------
// Kernel 3: W2 transpose -> W2t[n][k] (K contiguous) for panel staging.
// ---------------------------------------------------------------------------
__global__ void w2t_kernel(const float* __restrict__ W2, float* __restrict__ W2t) {
  int idx = blockIdx.x * blockDim.x + threadIdx.x;
  if (idx >= HD * NLAB) return;
  int k = idx >> 7;          // 0..1023
  int n = idx & (NLAB - 1);  // 0..127
  W2t[(size_t)n * HD + k] = W2[idx];
}

// ---------------------------------------------------------------------------
// Kernel 4: scores GEMM.  128 threads = 4 waves; each wave owns one 16-span
// M-tile; the block shares LDS-staged 128x128 W2 K-panels.  Per panel each
// wave stages its 16x128 ReLU'd hid panel, then runs 32 K-steps x 8 N-tiles
// of exact-fp32 V_WMMA_F32_16X16X4_F32 entirely out of LDS.
// Fragment layouts per CDNA5 ISA 7.12.2:
//   A 16x4 f32 : lane m=lane&15; VGPR0/1 hold K = 2*(lane>>4) + {0,1}
//   B 4x16 f32 : lane n=lane&15; VGPR0/1 hold K = 2*(lane>>4) + {0,1}
//   C/D 16x16  : VGPR v, lanes 0-15 -> row v, lanes 16-31 -> row v+8
// ---------------------------------------------------------------------------
__launch_bounds__(32 * WPB)
__global__ void scores_kernel(const float* __restrict__ PF, const float* __restrict__ PB,
                              const float* __restrict__ b1, const float* __restrict__ W2t,
                              const float* __restrict__ b2,
                              const int* __restrict__ spanL, const int* __restrict__ spanLeft,
                              float* __restrict__ S) {
  extern __shared__ float lds[];
  float* w2s = lds;                   // NLAB * PKP floats  (67,584 B)
  float* hids = lds + NLAB * PKP;     // WPB * 16 * PKP floats (33,792 B)

  const int tid = threadIdx.x;   // 0..127
  const int wave = tid >> 5;     // 0..3
  const int lane = tid & 31;
  const int mlane = lane & 15;
  const int khalf = lane >> 4;
  const int tile = blockIdx.x * WPB + wave;

  v8f acc[8];
  const v8f vzero = {0.f, 0.f, 0.f, 0.f, 0.f, 0.f, 0.f, 0.f};
#pragma unroll
  for (int nt = 0; nt < 8; ++nt) acc[nt] = vzero;

  for (int kc = 0; kc < HD / PK; ++kc) {
    __syncthreads(); // previous panel fully consumed before overwrite

    // ---- stage W2 panel (all 128 threads, float4, aligned: PKP*4=528=16*33)
    {
      int n = tid >> 5;             // 0..3 base, step 4
      int kq = (tid & 31) << 2;     // 0..124 step 4
      for (; n < NLAB; n += 4) {
        const float4 v = *(const float4*)(W2t + (size_t)n * HD + kc * PK + kq);
        *(float4*)(&w2s[n * PKP + kq]) = v;
      }
    }

    // ---- stage this wave's hid panel: relu(PF[r]-PF[l]+PB[l+1]-PB[r+1]+b1)
    for (int m = 0; m < 16; ++m) {
      int s = tile * 16 + m;
      int l = spanL[s];
      int lf = spanLeft[s];
      int r = lf + l;
      const float* pfr = PF + (size_t)r * HD + kc * PK;
      const float* pfl = PF + (size_t)lf * HD + kc * PK;
      const float* pbl = PB + (size_t)(lf + 1) * HD + kc * PK;
      const float* pbr = PB + (size_t)(r + 1) * HD + kc * PK;
      const float* bb1 = b1 + kc * PK;
      float* hrow = &hids[(wave * 16 + m) * PKP];
#pragma unroll
      for (int j = lane; j < PK; j += 32) {
        float v = pfr[j] - pfl[j] + pbl[j] - pbr[j] + bb1[j];
        hrow[j] = v > 0.f ? v : 0.f;
      }
    }
    __syncthreads();

    // ---- WMMA out of LDS: 32 K-steps x 8 independent N-tile chains
    const float* ha = &hids[(wave * 16 + mlane) * PKP];
    for (int kk = 0; kk < PK / 4; ++kk) {
      int kb = kk * 4 + khalf * 2;
      v2f a;
      a.x = ha[kb + 0];
      a.y = ha[kb + 1];
#pragma unroll
      for (int nt = 0; nt < 8; ++nt) {
        const float* wb = &w2s[(nt * 16 + mlane) * PKP];
        v2f b;
        b.x = wb[kb + 0];
        b.y = wb[kb + 1];
        acc[nt] = __builtin_amdgcn_wmma_f32_16x16x4_f32(
            /*neg_a=*/false, a, /*neg_b=*/false, b,
            /*c_mod=*/(short)0, acc[nt], /*reuse_a=*/false, /*reuse_b=*/false);
      }
    }
  }

  // ---- epilogue: +b2, max over 128 labels, write S triangle
  float bb[8];
#pragma unroll
  for (int nt = 0; nt < 8; ++nt) bb[nt] = b2[nt * 16 + mlane];

#pragma unroll
  for (int v = 0; v < 8; ++v) {
    float mx = -3.0e38f;
#pragma unroll
    for (int nt = 0; nt < 8; ++nt) mx = fmaxf(mx, acc[nt][v] + bb[nt]);
    mx = fmaxf(mx, __shfl_xor(mx, 1, 32));
    mx = fmaxf(mx, __shfl_xor(mx, 2, 32));
    mx = fmaxf(mx, __shfl_xor(mx, 4, 32));
    mx = fmaxf(mx, __shfl_xor(mx, 8, 32));
    if (mlane == 0) S[tile * 16 + khalf * 8 + v] = mx;
  }
}

// ---------------------------------------------------------------------------
// Kernel 5: CKY max-plus DP.  Single 384-thread workgroup; B in global ws;
// __syncthreads() per length gives block-wide visibility of prior writes.
// comb[left] = max_{k=1..l-1} B[k][left] + B[l-k][left+k]
// ---------------------------------------------------------------------------
__launch_bounds__(384)
__global__ void dp_kernel(const float* __restrict__ S, float* __restrict__ B,
                          float* __restrict__ out) {
  __shared__ int offs[L_SEQ + 1];
  int tid = threadIdx.x;
  if (tid < L_SEQ) {
    int l = tid + 1;
    offs[l] = (l - 1) * L_SEQ - ((l - 1) * (l - 2)) / 2;
  }
  for (int i = tid; i < L_SEQ; i += 384) B[i] = S[i]; // row l=1
  __syncthreads();

  for (int l = 2; l <= L_SEQ; ++l) {
    int N = L_SEQ - l + 1;
    if (tid < N) {
      int left = tid;
      float comb = -3.0e38f;
      int k = 1;
      for (; k + 1 < l; k += 2) {
        float x0 = B[offs[k] + left] + B[offs[l - k] + left + k];
        float x1 = B[offs[k + 1] + left] + B[offs[l - k - 1] + left + k + 1];
        comb = fmaxf(comb, fmaxf(x0, x1));
      }
      for (; k < l; ++k)
        comb = fmaxf(comb, B[offs[k] + left] + B[offs[l - k] + left + k]);
      int ol = offs[l];
      B[ol + left] = S[ol + left] + comb;
    }
    __syncthreads();
  }
  if (tid == 0) out[0] = B[NSPANS - 1]; // B[L][0] = last triangle element
}

// ---------------------------------------------------------------------------
extern "C" void kernel_launch(void* const* d_in, const int* in_sizes, int n_in,
                              void* d_out, int out_size, void* d_ws, size_t ws_size,
                              hipStream_t stream) {
  const float* hiddens = (const float*)d_in[0]; // (386, 1024)
  const float* W1 = (const float*)d_in[1];      // (1024, 1024)
  const float* b1 = (const float*)d_in[2];      // (1024,)
  const float* W2 = (const float*)d_in[3];      // (1024, 128)
  const float* b2 = (const float*)d_in[4];      // (128,)
  // d_in[5] = seq_len scalar (fixed 384)

  float* fw = (float*)d_ws;
  float* PF = fw;                               // (L+2)*1024
  float* PB = PF + (size_t)(L_SEQ + 2) * HD;    // (L+2)*1024
  float* S = PB + (size_t)(L_SEQ + 2) * HD;     // NSPANS
  float* B = S + NSPANS;                        // NSPANS
  float* W2t = B + NSPANS;                      // 128*1024
  int* spanL = (int*)(W2t + (size_t)HD * NLAB); // NSPANS
  int* spanLeft = spanL + NSPANS;               // NSPANS

  const size_t scores_lds = (size_t)(NLAB * PKP + WPB * 16 * PKP) * sizeof(float); // 101,376 B

  spans_init_kernel<<<2, 256, 0, stream>>>(spanL, spanLeft);
  proj_kernel<<<(2 * (L_SEQ + 2) * HD) / 256, 256, 0, stream>>>(hiddens, W1, PF, PB);
  w2t_kernel<<<(HD * NLAB) / 256, 256, 0, stream>>>(W2, W2t);
  scores_kernel<<<NBLK, 32 * WPB, scores_lds, stream>>>(PF, PB, b1, W2t, b2, spanL,
                                                        spanLeft, S);
  dp_kernel<<<1, 384, 0, stream>>>(S, B, (float*)d_out);
}